// MultiHeadAttention_29317446762844
// MI455X (gfx1250) — compile-verified
//
#include <hip/hip_runtime.h>
#include <hip/hip_bf16.h>

// Problem constants (match reference)
#define BB 4
#define NN 2048
#define DD 256
#define HH 8
#define HDv 32
#define EE 65536
#define MASK_WORDS (NN / 32) // 64 u32 words per row

typedef float v2f __attribute__((ext_vector_type(2)));
typedef float v8f __attribute__((ext_vector_type(8)));
typedef int v4i __attribute__((ext_vector_type(4)));

#if defined(__AMDGCN__) && __has_builtin(__builtin_amdgcn_global_load_async_to_lds_b128)
#define HAVE_ASYNC_LDS 1
typedef __attribute__((address_space(1))) v4i* async_gptr;
typedef __attribute__((address_space(3))) v4i* async_lptr;
#else
#define HAVE_ASYNC_LDS 0
#endif

__device__ __forceinline__ v8f wmma4(v2f a, v2f b, v8f c) {
    // D = A(16x4 f32) * B(4x16 f32) + C(16x16 f32)
    return __builtin_amdgcn_wmma_f32_16x16x4_f32(false, a, false, b, (short)0, c,
                                                 false, false);
}

// ---------------------------------------------------------------- utilities
__global__ __launch_bounds__(256) void zero_f32_kernel(float* p, size_t n) {
    size_t i = (size_t)blockIdx.x * blockDim.x + threadIdx.x;
    size_t stride = (size_t)gridDim.x * blockDim.x;
    for (; i < n; i += stride) p[i] = 0.0f;
}

__global__ __launch_bounds__(256) void mask_zero_kernel(unsigned* mask) {
    int total = NN * MASK_WORDS;
    for (int i = blockIdx.x * blockDim.x + threadIdx.x; i < total;
         i += gridDim.x * blockDim.x)
        mask[i] = 0u;
}

__global__ __launch_bounds__(256) void mask_diag_kernel(unsigned* mask) {
    int i = blockIdx.x * blockDim.x + threadIdx.x;
    if (i < NN) atomicOr(&mask[(size_t)i * MASK_WORDS + (i >> 5)], 1u << (i & 31));
}

__global__ __launch_bounds__(256) void mask_scatter_kernel(const int* __restrict__ edge,
                                                           unsigned* mask) {
    int e = blockIdx.x * blockDim.x + threadIdx.x;
    if (e < EE) {
        int s = edge[e];
        int d = edge[EE + e];
        atomicOr(&mask[(size_t)s * MASK_WORDS + (d >> 5)], 1u << (d & 31));
        atomicOr(&mask[(size_t)d * MASK_WORDS + (s >> 5)], 1u << (s & 31));
    }
}

// ---------------------------------------------------------------- GEMM: Out = X * W^T + bias
// X: [M x 256] row-major, W: [256 x 256] row-major (out col n uses W row n).
// mode 0: scatter into QKV layout [B,H,N,32]; mode 1: write [M x 256] row-major.
// Block: 256 threads (8 waves), computes a 64x64 output tile.
// Tiles staged straight into LDS with GLOBAL_LOAD_ASYNC_TO_LDS_B128 when available.
__global__ __launch_bounds__(256) void gemm_xwT_kernel(const float* __restrict__ X,
                                                       const float* __restrict__ W,
                                                       const float* __restrict__ bias,
                                                       float* __restrict__ Out,
                                                       int mode) {
    __shared__ float As[64][36]; // X tile: [row][k]   (36-float pad -> distinct banks)
    __shared__ float Ws[64][36]; // W tile: [n ][k]    (un-transposed; B frag = contiguous f2)

    const int tid = threadIdx.x;
    const int lane = tid & 31;
    const int wave = tid >> 5;
    const int ln = lane & 15;
    const int koff = (lane < 16) ? 0 : 2;

    const int m0 = blockIdx.x * 64;
    const int n0 = blockIdx.y * 64;

    const int sr = wave & 3;         // 16-row strip within the 64-row tile
    const int cg = (wave >> 2) * 32; // 32-col group (two 16x16 subtiles)

    v8f c0 = {};
    v8f c1 = {};

    const int ldr = tid >> 2; // 0..63 : row loaded by this thread
    const int ldp = tid & 3;  // 0..3  : 8-float slice within the 32-k chunk

    for (int kk = 0; kk < 256; kk += 32) {
        // ---- stage X tile and W tile into LDS
        {
            const float* gA = X + (size_t)(m0 + ldr) * 256 + kk + ldp * 8;
            const float* gW = W + (size_t)(n0 + ldr) * 256 + kk + ldp * 8;
#if HAVE_ASYNC_LDS
            // CDNA5 async DMA: memory -> LDS without touching VGPRs (ASYNCcnt)
            __builtin_amdgcn_global_load_async_to_lds_b128(
                (async_gptr)gA, (async_lptr)&As[ldr][ldp * 8], 0, 0);
            __builtin_amdgcn_global_load_async_to_lds_b128(
                (async_gptr)gA, (async_lptr)&As[ldr][ldp * 8], 16, 0);
            __builtin_amdgcn_global_load_async_to_lds_b128(
                (async_gptr)gW, (async_lptr)&Ws[ldr][ldp * 8], 0, 0);
            __builtin_amdgcn_global_load_async_to_lds_b128(
                (async_gptr)gW, (async_lptr)&Ws[ldr][ldp * 8], 16, 0);
            asm volatile("s_wait_asynccnt 0x0" ::: "memory");
#else
            const float4* ga = (const float4*)gA;
            float4 a0 = ga[0], a1 = ga[1];
            As[ldr][ldp * 8 + 0] = a0.x; As[ldr][ldp * 8 + 1] = a0.y;
            As[ldr][ldp * 8 + 2] = a0.z; As[ldr][ldp * 8 + 3] = a0.w;
            As[ldr][ldp * 8 + 4] = a1.x; As[ldr][ldp * 8 + 5] = a1.y;
            As[ldr][ldp * 8 + 6] = a1.z; As[ldr][ldp * 8 + 7] = a1.w;

            const float4* gw = (const float4*)gW;
            float4 w0 = gw[0], w1 = gw[1];
            Ws[ldr][ldp * 8 + 0] = w0.x; Ws[ldr][ldp * 8 + 1] = w0.y;
            Ws[ldr][ldp * 8 + 2] = w0.z; Ws[ldr][ldp * 8 + 3] = w0.w;
            Ws[ldr][ldp * 8 + 4] = w1.x; Ws[ldr][ldp * 8 + 5] = w1.y;
            Ws[ldr][ldp * 8 + 6] = w1.z; Ws[ldr][ldp * 8 + 7] = w1.w;
#endif
            if (kk < 224) { // prefetch next k-chunk (global_prefetch_b8)
                __builtin_prefetch(gA + 32, 0, 1);
                __builtin_prefetch(gW + 32, 0, 1);
            }
        }
        __syncthreads();

        // ---- 8 k-chunks of 4, two 16x16 subtiles per wave -> 16 WMMAs per kk
#pragma unroll
        for (int ch = 0; ch < 8; ++ch) {
            const int k = ch * 4 + koff;
            v2f a = {As[sr * 16 + ln][k], As[sr * 16 + ln][k + 1]};
            v2f b0 = {Ws[cg + ln][k], Ws[cg + ln][k + 1]};
            v2f b1 = {Ws[cg + 16 + ln][k], Ws[cg + 16 + ln][k + 1]};
            c0 = wmma4(a, b0, c0);
            c1 = wmma4(a, b1, c1);
        }
        __syncthreads();
    }

    // ---- epilogue: C layout -> (m = r + (lane>=16 ? 8:0), n = ln)
    const int n_a = n0 + cg + ln;
    const int n_b = n_a + 16;
    const float ba = bias[n_a];
    const float bb_ = bias[n_b];
#pragma unroll
    for (int r = 0; r < 8; ++r) {
        const int m = m0 + sr * 16 + r + ((lane < 16) ? 0 : 8);
        const float va = c0[r] + ba;
        const float vb = c1[r] + bb_;
        if (mode == 0) {
            const int bidx = m >> 11;      // m / N
            const int node = m & (NN - 1); // m % N
            const int ha = n_a >> 5, da = n_a & 31;
            const int hb = n_b >> 5, db = n_b & 31;
            Out[(((size_t)bidx * HH + ha) * NN + node) * HDv + da] = va;
            Out[(((size_t)bidx * HH + hb) * NN + node) * HDv + db] = vb;
        } else {
            Out[(size_t)m * DD + n_a] = va;
            Out[(size_t)m * DD + n_b] = vb;
        }
    }
}

// ---------------------------------------------------------------- attention
// One block per (b, h, 16-row tile). Full 16x2048 score strip lives in LDS
// (CDNA5: 320 KB per WGP). Softmax + masked scores never touch HBM; mean
// probabilities accumulated with global f32 atomics.
#define S_STRIDE 2056 // 2048 + 8 pad floats -> conflict-free column reads

__global__ __launch_bounds__(256) void attention_kernel(
    const float* __restrict__ Q, const float* __restrict__ K,
    const float* __restrict__ V, const unsigned* __restrict__ mask,
    const float* __restrict__ temp, float* __restrict__ attended,
    float* __restrict__ attn_mean) {
    extern __shared__ float smem[];
    float* S = smem;                   // [16][S_STRIDE]
    float* acc = smem + 16 * S_STRIDE; // [16][32]

    const int tid = threadIdx.x;
    const int lane = tid & 31;
    const int wave = tid >> 5;
    const int ln = lane & 15;
    const int koff = (lane < 16) ? 0 : 2;

    const int row0 = blockIdx.x * 16;
    const int h = blockIdx.y;
    const int b = blockIdx.z;

    const float inv_t = 1.0f / temp[0];
    const float* Qbh = Q + ((size_t)b * HH + h) * NN * HDv;
    const float* Kbh = K + ((size_t)b * HH + h) * NN * HDv;
    const float* Vbh = V + ((size_t)b * HH + h) * NN * HDv;

    // Q tile (16x32) fragments in registers -- reused for all 128 column tiles
    v2f aq[8];
#pragma unroll
    for (int ch = 0; ch < 8; ++ch) {
        const float* qp = Qbh + (size_t)(row0 + ln) * HDv + ch * 4 + koff;
        aq[ch].x = qp[0];
        aq[ch].y = qp[1];
    }

    // ---- scores: each wave computes 16 column tiles (16 cols each)
    for (int t = 0; t < 16; ++t) {
        const int j = wave * 16 + t; // column tile index 0..127
        const int coln = j * 16 + ln;
        const float* kp = Kbh + (size_t)coln * HDv;
        v8f c = {};
#pragma unroll
        for (int ch = 0; ch < 8; ++ch) {
            v2f bf = {kp[ch * 4 + koff], kp[ch * 4 + koff + 1]};
            c = wmma4(aq[ch], bf, c);
        }
#pragma unroll
        for (int r = 0; r < 8; ++r) {
            const int m = r + ((lane < 16) ? 0 : 8);
            const int rown = row0 + m;
            const unsigned mw = mask[(size_t)rown * MASK_WORDS + (coln >> 5)];
            const bool conn = (mw >> (coln & 31)) & 1u;
            S[m * S_STRIDE + coln] = conn ? (c[r] * inv_t) : -__builtin_inff();
        }
    }
    __syncthreads();

    // ---- masked softmax per row (16 threads per row, wave32 shfl reductions)
    {
        const int row = tid >> 4;
        const int c0i = tid & 15;
        float mx = -__builtin_inff();
        for (int c = c0i; c < NN; c += 16) mx = fmaxf(mx, S[row * S_STRIDE + c]);
#pragma unroll
        for (int o = 8; o >= 1; o >>= 1) mx = fmaxf(mx, __shfl_xor(mx, o, 32));

        float sum = 0.0f;
        for (int c = c0i; c < NN; c += 16) {
            const float p = __expf(S[row * S_STRIDE + c] - mx);
            S[row * S_STRIDE + c] = p;
            sum += p;
        }
#pragma unroll
        for (int o = 8; o >= 1; o >>= 1) sum += __shfl_xor(sum, o, 32);
        const float inv = 1.0f / sum;

        float* amrow = attn_mean + ((size_t)b * NN + (row0 + row)) * NN;
        for (int c = c0i; c < NN; c += 16) {
            const float p = S[row * S_STRIDE + c] * inv;
            S[row * S_STRIDE + c] = p;
            atomicAdd(&amrow[c], p * (1.0f / HH)); // head-mean accumulation
        }
    }

    for (int i = tid; i < 16 * 32; i += blockDim.x) acc[i] = 0.0f;
    __syncthreads();

    // ---- attended = S(16x2048) @ V(2048x32); K-range split across 8 waves
    {
        v8f c0 = {}, c1 = {};
        const int kbase = wave * 256;
#pragma unroll 4
        for (int ch = 0; ch < 64; ++ch) {
            const int k = kbase + ch * 4 + koff;
            v2f a = {S[ln * S_STRIDE + k], S[ln * S_STRIDE + k + 1]};
            const float* vp = Vbh + (size_t)k * HDv;
            v2f b0 = {vp[ln], vp[HDv + ln]};
            v2f b1 = {vp[16 + ln], vp[HDv + 16 + ln]};
            c0 = wmma4(a, b0, c0);
            c1 = wmma4(a, b1, c1);
        }
#pragma unroll
        for (int r = 0; r < 8; ++r) {
            const int m = r + ((lane < 16) ? 0 : 8);
            atomicAdd(&acc[m * 32 + ln], c0[r]); // ds_add_f32 cross-wave reduce
            atomicAdd(&acc[m * 32 + 16 + ln], c1[r]);
        }
    }
    __syncthreads();

    // ---- write attended tile [B,N,D], column block h*32..h*32+31
    for (int i = tid; i < 16 * 32; i += blockDim.x) {
        const int m = i >> 5;
        const int hd = i & 31;
        attended[((size_t)b * NN + row0 + m) * DD + h * HDv + hd] = acc[i];
    }
}

// ---------------------------------------------------------------- launch
extern "C" void kernel_launch(void* const* d_in, const int* in_sizes, int n_in,
                              void* d_out, int out_size, void* d_ws, size_t ws_size,
                              hipStream_t stream) {
    const float* x = (const float*)d_in[0];
    const int* edge = (const int*)d_in[1];
    const float* Wq = (const float*)d_in[2];
    const float* bq = (const float*)d_in[3];
    const float* Wk = (const float*)d_in[4];
    const float* bk = (const float*)d_in[5];
    const float* Wv = (const float*)d_in[6];
    const float* bv = (const float*)d_in[7];
    const float* Wo = (const float*)d_in[8];
    const float* bo = (const float*)d_in[9];
    const float* temp = (const float*)d_in[10];

    float* out = (float*)d_out;                    // [B,N,D]
    float* attn_mean = out + (size_t)BB * NN * DD; // [B,N,N]

    // workspace layout (mask words alias the first 131072 "floats")
    unsigned* mask = (unsigned*)d_ws;
    float* F = (float*)d_ws;
    const size_t QKV_ELEMS = (size_t)BB * HH * NN * HDv; // 2,097,152
    float* Qb = F + (size_t)NN * MASK_WORDS;
    float* Kb = Qb + QKV_ELEMS;
    float* Vb = Kb + QKV_ELEMS;
    float* Att = Vb + QKV_ELEMS;

    // 1) zero the attn-mean output region (we accumulate into it atomically)
    zero_f32_kernel<<<2048, 256, 0, stream>>>(attn_mean, (size_t)BB * NN * NN);

    // 2) adjacency bitmask: zero, self-loops, symmetric edge scatter
    mask_zero_kernel<<<512, 256, 0, stream>>>(mask);
    mask_diag_kernel<<<(NN + 255) / 256, 256, 0, stream>>>(mask);
    mask_scatter_kernel<<<EE / 256, 256, 0, stream>>>(edge, mask);

    // 3) Q/K/V projections (WMMA f32 GEMM, scatter to [B,H,N,32])
    dim3 gproj(8192 / 64, 256 / 64, 1);
    gemm_xwT_kernel<<<gproj, 256, 0, stream>>>(x, Wq, bq, Qb, 0);
    gemm_xwT_kernel<<<gproj, 256, 0, stream>>>(x, Wk, bk, Kb, 0);
    gemm_xwT_kernel<<<gproj, 256, 0, stream>>>(x, Wv, bv, Vb, 0);

    // 4) attention: 128 row-tiles x 8 heads x 4 batches; ~130 KB dynamic LDS
    const size_t att_lds = (size_t)(16 * S_STRIDE + 16 * 32) * sizeof(float);
    attention_kernel<<<dim3(NN / 16, HH, BB), 256, att_lds, stream>>>(
        Qb, Kb, Vb, mask, temp, Att, attn_mean);

    // 5) output projection -> d_out[0 .. B*N*D)
    gemm_xwT_kernel<<<gproj, 256, 0, stream>>>(Att, Wo, bo, out, 1);
}